// Brownian_OU_64785286693488
// MI455X (gfx1250) — compile-verified
//
#include <hip/hip_runtime.h>

// Problem constants from the reference
#define NB    256
#define NT    2048
#define NDIM  64
#define NTILE (NT / 16)   // 128 time tiles of 16
#define NBUF  16          // async LDS ring depth (tiles); wait = 2*(NBUF-1) = 30

typedef __attribute__((ext_vector_type(2))) float v2f;
typedef __attribute__((ext_vector_type(8))) float v8f;

// One wave owns one (batch b, 16-dim tile d0) slab and scans T.
//  - noise tiles stream into a private 16-tile LDS ring via
//    global_load_async_to_lds_b128 (ASYNCcnt): ~16KB in flight per wave,
//    ~16MB across the 1024-wave grid -> enough to feed 23.3 TB/s HBM.
//  - cumsum over each 16-step tile = L(16x16 lower-triangular ones) @ r via
//    4 chained V_WMMA_F32_16X16X4_F32; running carry broadcast through C,
//    re-extracted from row M=15 (VGPR7, lanes 16-31).
//  - per-row std multiplier and normalization factor computed once per wave
//    (lane n owns row n), redistributed with ds_bpermute shuffles.
//  - OU vs BM is a compile-time template selected by a *scalar* (blockIdx)
//    branch, so the loop body has no exec-mask divergence at all.
template <bool OU>
__device__ __forceinline__ void scan_slab(const float* __restrict__ tsb,
                                          const float* __restrict__ nb,
                                          float* __restrict__ ob,
                                          float* __restrict__ waveRing,
                                          int d0, int lane)
{
  const int q = lane >> 4;                // half-wave id
  const int n = lane & 15;                // column in tile / owned row index
  const int c = d0 + n;

  // ---- loop-invariant triangular A chunks (f32 16x4 layout) ----
  v2f A[4];
#pragma unroll
  for (int k = 0; k < 4; ++k) {
    const int k0 = 4 * k + 2 * q;
    A[k].x = (n >= k0)     ? 1.0f : 0.0f;
    A[k].y = (n >= k0 + 1) ? 1.0f : 0.0f;
  }

  // ---- per-lane offsets for async b128 tile copies ----
  // instr0 covers tile rows 0-7 (lane: row=lane>>2, 4-float group=lane&3),
  // instr1 covers rows 8-15 (+2048B global, +512B LDS).
  const uint32_t laneL0 = (uint32_t)(((lane >> 2) * 16 + (lane & 3) * 4) * 4);
  const uint64_t laneG0 = (uint64_t)(((lane >> 2) * NDIM + (lane & 3) * 4) * 4);
  const uint32_t ldsBase = (uint32_t)(uintptr_t)waveRing;

  auto issue_tile = [&](int j) {
    const uint64_t g = (uint64_t)(uintptr_t)(nb + (size_t)j * 16 * NDIM + d0) + laneG0;
    const uint32_t l = ldsBase + (uint32_t)((j & (NBUF - 1)) * 1024) + laneL0;
    asm volatile("global_load_async_to_lds_b128 %0, %1, off"
                 :: "v"(l), "v"(g) : "memory");
    asm volatile("global_load_async_to_lds_b128 %0, %1, off"
                 :: "v"(l + 512u), "v"(g + 2048ull) : "memory");
  };

  // ---- prologue: fill the ring ----
#pragma unroll
  for (int j = 0; j < NBUF; ++j) issue_tile(j);

  const float OUS = 2.23606797749979f;    // SIGMA / sqrt(2*theta) = sqrt(5)
  float carry = 0.0f;
  float prevT = 0.0f;                     // ts[-1] := 0
  float prevE = 1.0f;                     // exp(0.2*ts[-1]) := 1
  float tcur  = tsb[n];                   // lane n owns row n of current tile

  for (int t = 0; t < NTILE; ++t) {
    const int t16 = t * 16;
    // software-pipelined ts load for the next tile (clamped at the end)
    const int nbase = (t16 + 16 < NT) ? (t16 + 16) : t16;
    const float tnext = tsb[nbase + n];

    // oldest ring tile is complete when <= 2*(NBUF-1) async ops pending
    asm volatile("s_wait_asynccnt 0x1e" ::: "memory");

    // ---- per-row multiplier (std*scale, zero-guarded) and norm factor ----
    float mult, norm;
    if (OU) {
      const float e    = __expf(0.2f * tcur);
      const float em1s = __shfl(e, (n - 1) & 15, 32);
      const float em1  = (n == 0) ? prevE : em1s;
      const float d    = __builtin_amdgcn_sqrtf(e - em1);
      mult = (d > 0.0f) ? d * OUS : 0.0f;
      const float einv = __builtin_amdgcn_rcpf(e);            // exp(-0.2 t)
      const float den  = __builtin_amdgcn_sqrtf((1.0f - einv) * 5.0f) + 1e-8f;
      norm = __builtin_amdgcn_rsqf(e) * __builtin_amdgcn_rcpf(den); // exp(-0.1t)/den
      prevE = __shfl(e, 15, 32);
    } else {
      const float tm1s = __shfl(tcur, (n - 1) & 15, 32);
      const float tm1  = (n == 0) ? prevT : tm1s;
      const float d    = __builtin_amdgcn_sqrtf(tcur - tm1);
      mult = (d > 0.0f) ? d : 0.0f;
      norm = __builtin_amdgcn_rcpf(__builtin_amdgcn_sqrtf(tcur) + 1e-8f);
      prevT = __shfl(tcur, 15, 32);
    }

    // ---- 16x16 cumsum tile: 4 chained f32 WMMAs over the LDS tile ----
    const float* tile = waveRing + (t & (NBUF - 1)) * 256;
    v8f acc;
#pragma unroll
    for (int i = 0; i < 8; ++i) acc[i] = carry;

#pragma unroll
    for (int k = 0; k < 4; ++k) {
      const int rA = 4 * k + 2 * q;             // B-layout: VGPR0 row, VGPR1 row+1
      const float nAv = tile[rA * 16 + n];
      const float nBv = tile[rA * 16 + 16 + n];
      const float mA  = __shfl(mult, rA, 32);
      const float mB  = __shfl(mult, rA + 1, 32);
      v2f Bm;
      Bm.x = mA * nAv;
      Bm.y = mB * nBv;
      acc = __builtin_amdgcn_wmma_f32_16x16x4_f32(
          false, A[k], false, Bm, (short)0, acc, false, false);
    }

    // new carry: D[15][col] -> VGPR7 of lane 16+col
    carry = __shfl(acc[7], 16 + n, 32);

    // refill the freed ring slot (its reads are consumed by the WMMAs above)
    if (t + NBUF < NTILE) issue_tile(t + NBUF);

    // ---- normalize + streaming stores (D-layout: lane, VGPR i -> row i+8q) ----
#pragma unroll
    for (int i = 0; i < 8; ++i) {
      const float f = __shfl(norm, i + 8 * q, 32);
      __builtin_nontemporal_store(acc[i] * f,
                                  &ob[(size_t)(t16 + i + 8 * q) * NDIM + c]);
    }

    tcur = tnext;
  }
}

// Grid: 512 blocks of 64 threads (2 waves). blockIdx encodes (batch, BM/OU half)
// so the BM-vs-OU split is a scalar (SGPR) branch -> no exec-mask divergence.
__global__ __launch_bounds__(64) void brownian_ou_wmma_async(
    const float* __restrict__ ts,     // [B, T]
    const float* __restrict__ noise,  // [B, T, 64]
    float* __restrict__ out)          // [B, T, 64]
{
  __shared__ float ring[2][NBUF * 256];   // 2 waves x 16 tiles x 1KB = 32KB

  const int b    = blockIdx.x >> 1;
  const int half = blockIdx.x & 1;        // 0 = Brownian dims 0-31, 1 = OU dims 32-63
  const int wave = threadIdx.x >> 5;      // 0..1 -> 16-dim tile within the half
  const int lane = threadIdx.x & 31;
  const int d0   = half * 32 + wave * 16;

  const float* tsb = ts    + (size_t)b * NT;
  const float* nb  = noise + (size_t)b * NT * NDIM;
  float*       ob  = out   + (size_t)b * NT * NDIM;
  float*       wr  = &ring[wave][0];

  if (half)
    scan_slab<true>(tsb, nb, ob, wr, d0, lane);
  else
    scan_slab<false>(tsb, nb, ob, wr, d0, lane);
}

extern "C" void kernel_launch(void* const* d_in, const int* in_sizes, int n_in,
                              void* d_out, int out_size, void* d_ws, size_t ws_size,
                              hipStream_t stream) {
  (void)in_sizes; (void)n_in; (void)out_size; (void)d_ws; (void)ws_size;
  const float* ts    = (const float*)d_in[0];   // [B,T,1]
  const float* noise = (const float*)d_in[1];   // [B,T,DIM]
  float*       out   = (float*)d_out;           // [B,T,DIM]
  brownian_ou_wmma_async<<<NB * 2, 64, 0, stream>>>(ts, noise, out);
}